// AmpLoss_824633721711
// MI455X (gfx1250) — compile-verified
//
#include <hip/hip_runtime.h>
#include <math.h>

// ---------------------------------------------------------------------------
// AMP blood-pressure loss: 24 global reductions over N elements + tiny epilog.
// Memory-bound: reads 4 * N * 4B = 64MB (mbp_pred/m are unused by the math).
// Strategy: grid-stride float4 streaming, branchless f32 per-thread partials,
// wave reduction via V_WMMA_F32_16X16X4_F32 (ones-matrix column-sum trick),
// LDS cross-wave combine, f64 global atomics, single-thread finalize.
// ---------------------------------------------------------------------------

typedef __attribute__((ext_vector_type(2))) float v2f;
typedef __attribute__((ext_vector_type(8))) float v8f;

#define NACC 24
// acc[] layout (doubles in d_ws):
//  0        : count(dbp_pred < sbp_pred)
//  1..5     : mask counts
//  6..10    : sum mask_i * smape_d
// 11..15    : sum mask_i * smape_s
// 16..19    : sum d, sum dp, sum d^2, sum dp^2
// 20..23    : sum s, sum sp, sum s^2, sum sp^2

// Full 32-lane sum using one WMMA + 4 xor-shuffles.
// A = ones(16x4), B holds each lane's value in one slot (b = {p, 0}), so
// D[m,n] = sum_k B[k,n]: the WMMA folds lanes n and n+16 into column n.
// Each lane's D[0] holds one column sum; xor-shuffles over the 16-lane half
// (which covers all 16 columns) complete the total. Result in every lane.
__device__ __forceinline__ float wave_sum32(float p) {
    v2f a; a[0] = 1.0f; a[1] = 1.0f;
    v2f b; b[0] = p;    b[1] = 0.0f;
    v8f c = {};
    v8f dm = __builtin_amdgcn_wmma_f32_16x16x4_f32(
        /*neg_a=*/false, a, /*neg_b=*/false, b,
        /*c_mod=*/(short)0, c, /*reuse_a=*/false, /*reuse_b=*/false);
    float r = dm[0];
    r += __shfl_xor(r, 1, 32);
    r += __shfl_xor(r, 2, 32);
    r += __shfl_xor(r, 4, 32);
    r += __shfl_xor(r, 8, 32);
    return r;
}

__global__ void __launch_bounds__(64)
amp_zero_acc(double* __restrict__ acc) {
    if (threadIdx.x < NACC) acc[threadIdx.x] = 0.0;
}

__global__ void __launch_bounds__(256)
amp_partial(const float* __restrict__ dp_, const float* __restrict__ sp_,
            const float* __restrict__ d_,  const float* __restrict__ s_,
            int n, double* __restrict__ acc) {
    float A[NACC];
#pragma unroll
    for (int j = 0; j < NACC; ++j) A[j] = 0.0f;

    auto proc = [&](float dp, float sp, float dd, float ss) {
        A[0] += (dp < sp) ? 1.0f : 0.0f;
        float smd = 2.0f * fabsf(dp - dd) / (fabsf(dp) + fabsf(dd));
        float sms = 2.0f * fabsf(sp - ss) / (fabsf(sp) + fabsf(ss));
        float m0 = ((ss < 120.0f) && (dd < 80.0f)) ? 1.0f : 0.0f;
        float m1 = ((ss >= 120.0f) && (ss < 130.0f) && (dd < 80.0f)) ? 1.0f : 0.0f;
        float m2 = (((ss >= 130.0f) && (ss < 140.0f)) ||
                    ((dd >= 80.0f) && (dd < 90.0f))) ? 1.0f : 0.0f;
        float m3 = ((ss >= 140.0f) || (dd >= 90.0f)) ? 1.0f : 0.0f;
        float m4 = ((ss > 180.0f) || (dd > 120.0f)) ? 1.0f : 0.0f;
        A[1] += m0; A[6]  += m0 * smd; A[11] += m0 * sms;
        A[2] += m1; A[7]  += m1 * smd; A[12] += m1 * sms;
        A[3] += m2; A[8]  += m2 * smd; A[13] += m2 * sms;
        A[4] += m3; A[9]  += m3 * smd; A[14] += m3 * sms;
        A[5] += m4; A[10] += m4 * smd; A[15] += m4 * sms;
        A[16] += dd; A[17] += dp; A[18] += dd * dd; A[19] += dp * dp;
        A[20] += ss; A[21] += sp; A[22] += ss * ss; A[23] += sp * sp;
    };

    const int tid      = blockIdx.x * blockDim.x + threadIdx.x;
    const int nthreads = gridDim.x * blockDim.x;
    const int n4       = n >> 2;

    const float4* dp4 = (const float4*)dp_;
    const float4* sp4 = (const float4*)sp_;
    const float4* dd4 = (const float4*)d_;
    const float4* ss4 = (const float4*)s_;

    for (int i = tid; i < n4; i += nthreads) {
        float4 vdp = dp4[i];
        float4 vsp = sp4[i];
        float4 vdd = dd4[i];
        float4 vss = ss4[i];
        proc(vdp.x, vsp.x, vdd.x, vss.x);
        proc(vdp.y, vsp.y, vdd.y, vss.y);
        proc(vdp.z, vsp.z, vdd.z, vss.z);
        proc(vdp.w, vsp.w, vdd.w, vss.w);
    }
    // Scalar tail (n not multiple of 4)
    const int base = n4 << 2;
    if (base + tid < n) proc(dp_[base + tid], sp_[base + tid],
                             d_[base + tid],  s_[base + tid]);

    // ---- wave reduction via WMMA (EXEC all-ones here: no divergence) ----
#pragma unroll
    for (int j = 0; j < NACC; ++j) A[j] = wave_sum32(A[j]);

    // ---- cross-wave combine in LDS, then one f64 atomic per accumulator ----
    __shared__ float red[8][NACC];
    const int lane = threadIdx.x & 31;
    const int wv   = threadIdx.x >> 5;
    if (lane == 0) {
#pragma unroll
        for (int j = 0; j < NACC; ++j) red[wv][j] = A[j];
    }
    __syncthreads();
    if (threadIdx.x < NACC) {
        float bsum = 0.0f;
        const int nw = blockDim.x >> 5;
        for (int w = 0; w < nw; ++w) bsum += red[w][threadIdx.x];
        atomicAdd(&acc[threadIdx.x], (double)bsum);
    }
}

__global__ void __launch_bounds__(32)
amp_finalize(const double* __restrict__ acc, float* __restrict__ out, int n) {
    if (threadIdx.x != 0) return;
    const double nf = (double)n;

    const double scale_cost = 1.0 - acc[0] / nf;

    double d_rst = 0.0, s_rst = 0.0, rst_d = 0.0, rst_s = 0.0;
    int cnt = 0;
    for (int i = 0; i < 5; ++i) {
        double cti  = acc[1 + i];
        double safe = (cti > 1.0) ? cti : 1.0;
        double w    = sqrt(log(nf / safe));
        double sum_d = w * acc[6 + i];
        double sum_s = w * acc[11 + i];
        double nd = (d_rst + sum_d) / safe;
        double ns = (s_rst + sum_s) / safe;
        if (cti > 0.0) { d_rst = nd; s_rst = ns; rst_d += nd; rst_s += ns; ++cnt; }
    }
    const double denom = (cnt == 0) ? 5.0 : (double)cnt;
    rst_d /= denom;
    rst_s /= denom;

    const double sd = acc[16], sdp = acc[17], sd2 = acc[18], sdp2 = acc[19];
    const double ss = acc[20], ssp = acc[21], ss2 = acc[22], ssp2 = acc[23];
    const double mean_d = sd / nf, mean_dp = sdp / nf;
    const double mean_s = ss / nf, mean_sp = ssp / nf;
    const double var_d  = (sd2  - sd  * sd  / nf) / (nf - 1.0);
    const double var_dp = (sdp2 - sdp * sdp / nf) / (nf - 1.0);
    const double var_s  = (ss2  - ss  * ss  / nf) / (nf - 1.0);
    const double var_sp = (ssp2 - ssp * ssp / nf) / (nf - 1.0);

    const double dbp_mean_cost = fabs(mean_d - mean_dp) / mean_d;
    const double sbp_mean_cost = fabs(mean_s - mean_sp) / mean_s;
    const double dbp_var_cost  = fabs(var_d - var_dp) / var_d;
    const double sbp_var_cost  = fabs(var_s - var_sp) / var_s;

    out[0] = (float)(rst_d + dbp_mean_cost + dbp_var_cost);
    out[1] = (float)(rst_s + sbp_mean_cost + sbp_var_cost);
    out[2] = (float)scale_cost;
}

extern "C" void kernel_launch(void* const* d_in, const int* in_sizes, int n_in,
                              void* d_out, int out_size, void* d_ws, size_t ws_size,
                              hipStream_t stream) {
    // Input order: dbp_pred, sbp_pred, mbp_pred, d, s, m (mbp_pred/m unused).
    const float* dp = (const float*)d_in[0];
    const float* sp = (const float*)d_in[1];
    const float* dd = (const float*)d_in[3];
    const float* ss = (const float*)d_in[4];
    const int n = in_sizes[3];

    double* acc = (double*)d_ws;   // 24 doubles = 192 bytes of scratch
    float*  out = (float*)d_out;

    amp_zero_acc<<<1, 64, 0, stream>>>(acc);

    const int block = 256;            // 8 wave32s / block
    int n4 = n >> 2;
    int grid = (n4 + block - 1) / block;
    if (grid > 1024) grid = 1024;     // 4 float4-iters/thread at N=4M
    if (grid < 1) grid = 1;
    amp_partial<<<grid, block, 0, stream>>>(dp, sp, dd, ss, n, acc);

    amp_finalize<<<1, 32, 0, stream>>>(acc, out, n);
}